// net_37228776522052
// MI455X (gfx1250) — compile-verified
//
#include <hip/hip_runtime.h>

// ---------------- problem constants ----------------
#define NBATCH  4u
#define NANCH   (1u << 20)            // 1048576 anchors
#define NROWS   (NBATCH * NANCH)      // 4194304 rows in probs/deltas
#define NFLOAT  (NROWS * 3u)          // 12582912 floats in probs
#define KSEL    6000u
#define FGMAX   0x7FFFFFu             // NFG-1, NFG = NROWS*2 = 2^23

// ---------------- tuning ----------------
#define SCAN_BLK  256                 // 8 waves (wave32)
#define TILE_F    1024                // floats staged per block-iteration (4KB)
#define SCAN_GRID 1024                // 12288 tiles total -> 12 per block
#define NBIN      4096
#define CAP       16384               // candidate buffer (>= KSEL with slack)

// ---------------- CDNA5 async global->LDS staging ----------------
typedef int v4i __attribute__((vector_size(16)));
typedef __attribute__((address_space(1))) v4i g_v4i;   // global (prints as __device__)
typedef __attribute__((address_space(3))) v4i l_v4i;   // LDS

#if defined(__has_builtin)
#if __has_builtin(__builtin_amdgcn_global_load_async_to_lds_b128)
#define HAVE_ASYNC_LDS 1
#endif
#endif

__device__ __forceinline__ void stage16(const float* g, float* l) {
#ifdef HAVE_ASYNC_LDS
    // CDNA5 async data path: global_load_async_to_lds_b128 (tracked by ASYNCcnt)
    __builtin_amdgcn_global_load_async_to_lds_b128(
        (g_v4i*)(unsigned long long)g,
        (l_v4i*)(unsigned)(unsigned long long)l,
        0, 0);
#else
    *(float4*)l = *(const float4*)g;
#endif
}

__device__ __forceinline__ void stage_wait_barrier() {
#ifdef HAVE_ASYNC_LDS
#if __has_builtin(__builtin_amdgcn_s_wait_asynccnt)
    __builtin_amdgcn_s_wait_asynccnt(0);
#else
    asm volatile("s_wait_asynccnt 0" ::: "memory");
#endif
#endif
    __syncthreads();
}

// ---------------- kernels ----------------
__global__ void zero_kernel(unsigned* p, int n) {
    int i = blockIdx.x * blockDim.x + threadIdx.x;
    if (i < n) p[i] = 0u;
}

// Pass 1: histogram of key bits [30:19] over all fg scores.
__global__ void hist1_kernel(const float* __restrict__ probs,
                             unsigned* __restrict__ ghist) {
    __shared__ float    tile[TILE_F];
    __shared__ unsigned lh[NBIN];
    const unsigned tid = threadIdx.x;
    for (unsigned i = tid; i < NBIN; i += SCAN_BLK) lh[i] = 0u;
    __syncthreads();
    for (unsigned base = blockIdx.x * TILE_F; base < NFLOAT;
         base += gridDim.x * TILE_F) {
        stage16(probs + base + tid * 4u, &tile[tid * 4u]);
        stage_wait_barrier();
#pragma unroll
        for (unsigned e = 0; e < 4u; ++e) {
            unsigned t = base + tid * 4u + e;
            unsigned r = t % 3u;                 // column within (bg, fg0, fg1)
            if (r != 0u) {
                unsigned key = __float_as_uint(tile[tid * 4u + e]);
                atomicAdd(&lh[(key >> 19) & (NBIN - 1u)], 1u);
            }
        }
        __syncthreads();                          // tile reuse
    }
    for (unsigned i = tid; i < NBIN; i += SCAN_BLK) {
        unsigned v = lh[i];
        if (v) atomicAdd(&ghist[i], v);
    }
}

// Find critical bin from the top.  pass==0: write P1/A1.  pass==1: write L2.
__global__ void __launch_bounds__(1024)
findcrit_kernel(const unsigned* __restrict__ hist, unsigned* params, int pass) {
    __shared__ unsigned part[1024];
    const unsigned tid = threadIdx.x;
    unsigned s = 0u;
#pragma unroll
    for (int i = 0; i < 4; ++i) s += hist[tid * 4u + i];
    part[tid] = s;
    __syncthreads();
    if (tid == 0) {
        unsigned A = (pass == 0) ? 0u : params[1];
        int seg = 1023;
        for (; seg >= 0; --seg) {
            if (A + part[seg] >= KSEL) break;
            A += part[seg];
        }
        if (seg < 0) seg = 0;
        int b = seg * 4 + 3;
        for (; b > seg * 4; --b) {
            unsigned c = hist[b];
            if (A + c >= KSEL) break;
            A += c;
        }
        if (pass == 0) { params[0] = (unsigned)b; params[1] = A; }
        else           { params[2] = (params[0] << 19) | ((unsigned)b << 7); }
    }
}

// Pass 2: refine histogram (bits [18:7]) among elements in critical bin P1.
__global__ void hist2_kernel(const float* __restrict__ probs,
                             unsigned* __restrict__ ghist,
                             const unsigned* __restrict__ params) {
    __shared__ float    tile[TILE_F];
    __shared__ unsigned lh[NBIN];
    const unsigned tid = threadIdx.x;
    const unsigned P1  = params[0];
    for (unsigned i = tid; i < NBIN; i += SCAN_BLK) lh[i] = 0u;
    __syncthreads();
    for (unsigned base = blockIdx.x * TILE_F; base < NFLOAT;
         base += gridDim.x * TILE_F) {
        stage16(probs + base + tid * 4u, &tile[tid * 4u]);
        stage_wait_barrier();
#pragma unroll
        for (unsigned e = 0; e < 4u; ++e) {
            unsigned t = base + tid * 4u + e;
            unsigned r = t % 3u;
            if (r != 0u) {
                unsigned key = __float_as_uint(tile[tid * 4u + e]);
                if ((key >> 19) == P1)
                    atomicAdd(&lh[(key >> 7) & (NBIN - 1u)], 1u);
            }
        }
        __syncthreads();
    }
    for (unsigned i = tid; i < NBIN; i += SCAN_BLK) {
        unsigned v = lh[i];
        if (v) atomicAdd(&ghist[i], v);
    }
}

// Pass 3: compact all candidates with key >= L2 as packed (key<<23)|(FGMAX-f).
__global__ void compact_kernel(const float* __restrict__ probs,
                               unsigned* __restrict__ params,
                               unsigned long long* __restrict__ cand) {
    __shared__ float tile[TILE_F];
    const unsigned tid = threadIdx.x;
    const unsigned L2v = params[2];
    for (unsigned base = blockIdx.x * TILE_F; base < NFLOAT;
         base += gridDim.x * TILE_F) {
        stage16(probs + base + tid * 4u, &tile[tid * 4u]);
        stage_wait_barrier();
#pragma unroll
        for (unsigned e = 0; e < 4u; ++e) {
            unsigned t = base + tid * 4u + e;
            unsigned r = t % 3u;
            if (r != 0u) {
                unsigned key = __float_as_uint(tile[tid * 4u + e]);
                if (key >= L2v) {
                    unsigned f   = (t / 3u) * 2u + (r - 1u);  // fg flat index
                    unsigned pos = atomicAdd(&params[3], 1u);
                    if (pos < CAP)
                        cand[pos] = ((unsigned long long)key << 23) |
                                    (unsigned long long)(FGMAX - f);
                }
            }
        }
        __syncthreads();
    }
}

// Final: sort candidates desc, exact top-K, per-group top-2, box math, output.
__global__ void __launch_bounds__(1024)
final_kernel(const unsigned long long* __restrict__ cand,
             const unsigned* __restrict__ params,
             const float* __restrict__ anchors,
             const float* __restrict__ deltas,
             float* __restrict__ out) {
    extern __shared__ unsigned long long srt[];      // CAP entries (128 KB)
    __shared__ int firstp[8], secondp[8], pos16[16];
    const unsigned tid = threadIdx.x;
    const unsigned NT  = blockDim.x;

    unsigned M = params[3];
    if (M > CAP) M = CAP;
    for (unsigned i = tid; i < CAP; i += NT) srt[i] = (i < M) ? cand[i] : 0ull;
    if (tid < 8u) { firstp[tid] = 0x7FFFFFFF; secondp[tid] = 0x7FFFFFFF; }
    __syncthreads();

    // bitonic sort, descending
    for (unsigned k = 2u; k <= CAP; k <<= 1) {
        for (unsigned j = k >> 1; j > 0u; j >>= 1) {
            for (unsigned i = tid; i < CAP; i += NT) {
                unsigned ixj = i ^ j;
                if (ixj > i) {
                    unsigned long long a = srt[i], b = srt[ixj];
                    bool up = ((i & k) == 0u);       // descending blocks
                    if (up ? (a < b) : (a > b)) { srt[i] = b; srt[ixj] = a; }
                }
            }
            __syncthreads();
        }
    }

    // positions of first/second entry per (batch,class) group inside top-K
    unsigned lim = (M < KSEL) ? M : KSEL;
    for (unsigned p = tid; p < lim; p += NT) {
        unsigned f = FGMAX - (unsigned)(srt[p] & 0x7FFFFFull);
        unsigned g = ((f >> 21) << 1) | (f & 1u);    // batch*2 + (cls-1)
        atomicMin(&firstp[g], (int)p);
    }
    __syncthreads();
    for (unsigned p = tid; p < lim; p += NT) {
        unsigned f = FGMAX - (unsigned)(srt[p] & 0x7FFFFFull);
        unsigned g = ((f >> 21) << 1) | (f & 1u);
        if ((int)p != firstp[g]) atomicMin(&secondp[g], (int)p);
    }
    __syncthreads();

    if (tid == 0) {
        int top = (int)lim - 1; if (top < 0) top = 0;
        for (int g = 0; g < 8; ++g) {
            int a = firstp[g];  if (a > top) a = top;
            int b = secondp[g]; if (b > top) b = top;
            pos16[g * 2] = a; pos16[g * 2 + 1] = b;
        }
        for (int i = 1; i < 16; ++i) {               // sort ascending
            int v = pos16[i], j = i - 1;
            while (j >= 0 && pos16[j] > v) { pos16[j + 1] = pos16[j]; --j; }
            pos16[j + 1] = v;
        }
    }
    __syncthreads();

    if (tid < 16u) {
        const float scl[6]  = {256.f, 256.f, 256.f, 256.f, 128.f, 128.f};
        const float stdv[6] = {0.1f, 0.1f, 0.1f, 0.2f, 0.2f, 0.2f};
        int p = pos16[tid];
        unsigned long long e = srt[p];
        unsigned key  = (unsigned)(e >> 23);
        float    scr  = __uint_as_float(key);
        unsigned f    = FGMAX - (unsigned)(e & 0x7FFFFFull);
        unsigned prop = f >> 1;
        unsigned cls  = (f & 1u) + 1u;
        unsigned bix  = prop >> 20;                  // NA = 2^20
        unsigned arow = prop & (NANCH - 1u);

        float anc[6], dlt[6];
#pragma unroll
        for (int j = 0; j < 6; ++j) {
            anc[j] = floorf(anchors[arow * 6u + j] / scl[j]);  // exact: pow2
            dlt[j] = deltas[(unsigned long long)prop * 6u + j] * stdv[j];
        }
        float h = anc[2] - anc[0], w = anc[3] - anc[1], d = anc[5] - anc[4];
        float cy = anc[0] + 0.5f * h + dlt[0] * h;
        float cx = anc[1] + 0.5f * w + dlt[1] * w;
        float cz = anc[4] + 0.5f * d + dlt[2] * d;
        h = h * expf(dlt[3]);
        w = w * expf(dlt[4]);
        d = d * expf(dlt[5]);
        float y1 = cy - 0.5f * h, x1 = cx - 0.5f * w, z1 = cz - 0.5f * d;
        float bx[6] = { y1, x1, y1 + h, x1 + w, z1, z1 + d };
        float* row = out + tid * 9u;
#pragma unroll
        for (int j = 0; j < 6; ++j) {
            float v = bx[j] * scl[j];
            v = fminf(fmaxf(v, 0.f), scl[j]);
            row[j] = rintf(v);                        // round-half-even = jnp.round
        }
        row[6] = (float)bix;
        row[7] = (float)cls;
        row[8] = scr;
    }
}

// ---------------- host launcher ----------------
extern "C" void kernel_launch(void* const* d_in, const int* in_sizes, int n_in,
                              void* d_out, int out_size, void* d_ws, size_t ws_size,
                              hipStream_t stream) {
    const float* anchors = (const float*)d_in[0];
    const float* probs   = (const float*)d_in[1];
    const float* deltas  = (const float*)d_in[2];
    (void)in_sizes; (void)n_in; (void)out_size; (void)ws_size;

    // workspace layout (needs ~164 KB):
    //   [0..4095]   hist1, [4096..8191] hist2, [8192..8207] params
    //   then CAP * 8 bytes of candidate buffer (8-byte aligned: offset 32832)
    unsigned* ws_u = (unsigned*)d_ws;
    unsigned* h1   = ws_u;
    unsigned* h2   = ws_u + NBIN;
    unsigned* pr   = ws_u + 2 * NBIN;                 // [0]=P1 [1]=A1 [2]=L2 [3]=count
    unsigned long long* cand = (unsigned long long*)(ws_u + 2 * NBIN + 16);

    const int nzero = 2 * NBIN + 16;
    zero_kernel<<<(nzero + 255) / 256, 256, 0, stream>>>(ws_u, nzero);
    hist1_kernel<<<SCAN_GRID, SCAN_BLK, 0, stream>>>(probs, h1);
    findcrit_kernel<<<1, 1024, 0, stream>>>(h1, pr, 0);
    hist2_kernel<<<SCAN_GRID, SCAN_BLK, 0, stream>>>(probs, h2, pr);
    findcrit_kernel<<<1, 1024, 0, stream>>>(h2, pr, 1);
    compact_kernel<<<SCAN_GRID, SCAN_BLK, 0, stream>>>(probs, pr, cand);
    final_kernel<<<1, 1024, CAP * sizeof(unsigned long long), stream>>>(
        cand, pr, anchors, deltas, (float*)d_out);
}